// AddRadiusEdgeIndex_59072980189797
// MI455X (gfx1250) — compile-verified
//
#include <hip/hip_runtime.h>

// radius_graph on MI455X (gfx1250, wave32).
//
// dist2[i,j] = |xi|^2 + |xj|^2 - 2 xi.xj is a K=4 f32 GEMM per 16x16 tile:
//   A[m,:] = ( x_i,  y_i,  z_i, |x_i|^2 )
//   B[:,n] = (-2x_j,-2y_j,-2z_j, 1     )
//   C[m,n] = |x_j|^2
// -> D = A*B + C = dist2 tile, one V_WMMA_F32_16X16X4_F32 per tile.
//
// Output (512 MB) is the bottleneck: ~22 us floor @ 23.3 TB/s. pos (96 KB)
// lives in L2 for the whole kernel, so tile re-reads are free. Output stores
// are non-temporal (TH=NT): write-once data larger than the 192 MB L2 should
// stream instead of displacing the (cache-worthy) pos array.

typedef float v2f __attribute__((ext_vector_type(2)));
typedef float v8f __attribute__((ext_vector_type(8)));

#define NPTS 8192
#define R2   1.0f

#define WAVES_PER_BLOCK 8
#define COLT 8   // column tiles per wave (A-fragment reuse factor)

__global__ __launch_bounds__(256) void AddRadiusEdgeIndex_kernel(
    const float* __restrict__ pos, float* __restrict__ out)
{
    const int lane = threadIdx.x & 31;
    const int wave = threadIdx.x >> 5;
    const int half = lane >> 4;     // 0: lanes 0-15, 1: lanes 16-31
    const int l16  = lane & 15;

    const int rowTile  = blockIdx.y;                                    // 0..511
    const int colTile0 = (blockIdx.x * WAVES_PER_BLOCK + wave) * COLT;  // 0..504

    // ---- A fragment: row i = rowTile*16 + (lane & 15)
    // 16x4 f32 A layout: lanes 0-15 -> {K0,K1}, lanes 16-31 -> {K2,K3}
    const int i = rowTile * 16 + l16;
    const float xi = pos[i * 3 + 0];
    const float yi = pos[i * 3 + 1];
    const float zi = pos[i * 3 + 2];
    const float sqi = xi * xi + yi * yi + zi * zi;
    v2f a;
    a.x = half ? zi  : xi;   // K=2 : K=0
    a.y = half ? sqi : yi;   // K=3 : K=1

    float* __restrict__ distOut = out;
    float* __restrict__ maskOut = out + (size_t)NPTS * (size_t)NPTS;

    #pragma unroll 2
    for (int t = 0; t < COLT; ++t) {
        const int colTile = colTile0 + t;
        const int j = colTile * 16 + l16;           // column n = l16
        const float xj = pos[j * 3 + 0];
        const float yj = pos[j * 3 + 1];
        const float zj = pos[j * 3 + 2];
        const float sqj = xj * xj + yj * yj + zj * zj;

        // 4x16 f32 B layout: VGPR v, lanes 0-15 -> K=v, lanes 16-31 -> K=v+2
        v2f b;
        b.x = half ? (-2.0f * zj) : (-2.0f * xj);   // K=2 : K=0
        b.y = half ? 1.0f         : (-2.0f * yj);   // K=3 : K=1

        // C[m,n] = sq_j: depends only on column n = l16 -> broadcast to all 8 VGPRs
        v8f c;
        #pragma unroll
        for (int v = 0; v < 8; ++v) c[v] = sqj;

        // D = A*B + C = dist2 tile (16x16 f32)
        v8f d = __builtin_amdgcn_wmma_f32_16x16x4_f32(
            /*neg_a=*/false, a, /*neg_b=*/false, b,
            /*c_mod=*/(short)0, c, /*reuse_a=*/false, /*reuse_b=*/false);

        // D layout: lane l, VGPR v -> (M = v + 8*half, N = l16)
        #pragma unroll
        for (int v = 0; v < 8; ++v) {
            const int row = rowTile * 16 + v + half * 8;
            const size_t idx = (size_t)row * NPTS + (size_t)colTile * 16 + l16;
            float dd = d[v] < 0.0f ? 0.0f : d[v];   // jnp.maximum(dist2, 0)
            const bool m = dd <= R2;                // edge_mask (loop=True incl.)
            // Streaming (non-temporal) stores: write-once output, bypass L2 retention.
            __builtin_nontemporal_store(m ? dd : 0.0f,   &distOut[idx]);  // masked_dist2
            __builtin_nontemporal_store(m ? 1.0f : 0.0f, &maskOut[idx]);  // edge_mask
        }
    }
}

extern "C" void kernel_launch(void* const* d_in, const int* in_sizes, int n_in,
                              void* d_out, int out_size, void* d_ws, size_t ws_size,
                              hipStream_t stream) {
    (void)in_sizes; (void)n_in; (void)out_size; (void)d_ws; (void)ws_size;
    const float* pos = (const float*)d_in[0];
    float* out = (float*)d_out;

    // 512x512 tiles of 16x16; each block: 16 rows x (8 waves * 8 tiles * 16) = 1024 cols
    dim3 block(256, 1, 1);
    dim3 grid(NPTS / 16 / (WAVES_PER_BLOCK * COLT), NPTS / 16, 1);  // (8, 512)
    AddRadiusEdgeIndex_kernel<<<grid, block, 0, stream>>>(pos, out);
}